// TransitionUp_44332652430157
// MI455X (gfx1250) — compile-verified
//
#include <hip/hip_runtime.h>
#include <hip/hip_bf16.h>

// ---------------------------------------------------------------------------
// Shapes fixed by the reference
// ---------------------------------------------------------------------------
#define BB   2
#define NN   2048
#define MM   8192
#define CIN  512
#define COUT 256
#define BN_EPS 1e-5f
#define NN_EPS 1e-8f

typedef __attribute__((ext_vector_type(2))) float v2f;
typedef __attribute__((ext_vector_type(8))) float v8f;

// async-to-LDS builtin wants pointers to 16-byte int vectors (per hipcc diag)
typedef int vsi4 __attribute__((vector_size(16)));
typedef __attribute__((address_space(1))) vsi4* gvec_p;
typedef __attribute__((address_space(3))) vsi4* lvec_p;

#if __has_builtin(__builtin_amdgcn_global_load_async_to_lds_b128)
#define HAVE_ASYNC_LDS 1
#else
#define HAVE_ASYNC_LDS 0
#endif

__device__ __forceinline__ v2f ld2(const float* p) { return *(const v2f*)p; }

#define WMMA_F32(A, Bv, C) \
    __builtin_amdgcn_wmma_f32_16x16x4_f32(false, (A), false, (Bv), (short)0, (C), false, false)

// ---------------------------------------------------------------------------
// Kernel 1: fp32 GEMM via V_WMMA_F32_16X16X4_F32, output TRANSPOSED:
//   Ht[b,p,o] = sum_k X[b,p,k] * W[o,k]     X:(B,P,K), W:(Cout,K), Ht:(B,P,Cout)
// One wave owns a 16(o) x 64(p) strip = four 16x16 tiles sharing the A
// fragment (5 loads : 4 WMMAs per K-step), with one-deep prefetch so the
// next K-slice's loads are in flight while the current WMMAs execute.
// A (16x4 f32) layout: VGPR0 = K0 (lanes0-15)/K2 (lanes16-31), VGPR1 = K1/K3.
// D layout: VGPR i = row (i + 8*(lane>=16)), col = lane&15  ->  per-lane
// contiguous 8-float store in the transposed output.
// ---------------------------------------------------------------------------
template <int K>
__global__ void gemm_wmma(const float* __restrict__ X,
                          const float* __restrict__ W,
                          float* __restrict__ Ht,
                          int P, int Cout)
{
    const int lane  = threadIdx.x & 31;
    const int wave  = threadIdx.x >> 5;
    const int gwave = blockIdx.x * (blockDim.x >> 5) + wave;

    const int tilesN4 = P >> 6;                 // groups of 4 n-tiles
    const int tiles_m = Cout >> 4;
    const int total   = BB * tiles_m * tilesN4;
    if (gwave >= total) return;                 // uniform per wave: EXEC all-1s

    const int nt4 = gwave % tilesN4;
    const int t2  = gwave / tilesN4;
    const int mt  = t2 % tiles_m;
    const int bb  = t2 / tiles_m;

    const int row = lane & 15;
    const int khi = lane >> 4;                  // 0: K0/K1 slot, 1: K2/K3 slot

    const float* Wrow = W + (size_t)(mt * 16 + row) * K + 2 * khi;
    const float* Xb   = X + ((size_t)bb * P + nt4 * 64 + row) * K + 2 * khi;
    const float* X0 = Xb;
    const float* X1 = Xb + 16 * (size_t)K;
    const float* X2 = Xb + 32 * (size_t)K;
    const float* X3 = Xb + 48 * (size_t)K;

    v8f acc0 = {}, acc1 = {}, acc2 = {}, acc3 = {};

    v2f aC  = ld2(Wrow);
    v2f b0C = ld2(X0), b1C = ld2(X1), b2C = ld2(X2), b3C = ld2(X3);

#pragma unroll 4
    for (int k = 4; k <= K; k += 4) {
        v2f aN, b0N, b1N, b2N, b3N;
        if (k < K) {                            // prefetch next K-slice
            aN  = ld2(Wrow + k);
            b0N = ld2(X0 + k);
            b1N = ld2(X1 + k);
            b2N = ld2(X2 + k);
            b3N = ld2(X3 + k);
        }
        acc0 = WMMA_F32(aC, b0C, acc0);
        acc1 = WMMA_F32(aC, b1C, acc1);
        acc2 = WMMA_F32(aC, b2C, acc2);
        acc3 = WMMA_F32(aC, b3C, acc3);
        aC = aN; b0C = b0N; b1C = b1N; b2C = b2N; b3C = b3N;
    }

    // Transposed store: Ht[(b*P + n), o] -- per lane 8 contiguous floats/tile
    const int nn    = lane & 15;
    const int mbase = khi * 8;
    float* Hb = Ht + ((size_t)bb * P + (size_t)nt4 * 64 + nn) * Cout
                   + mt * 16 + mbase;
#pragma unroll
    for (int i = 0; i < 8; ++i) Hb[(size_t)0 * 16 * Cout + i] = acc0[i];
#pragma unroll
    for (int i = 0; i < 8; ++i) Hb[(size_t)1 * 16 * Cout + i] = acc1[i];
#pragma unroll
    for (int i = 0; i < 8; ++i) Hb[(size_t)2 * 16 * Cout + i] = acc2[i];
#pragma unroll
    for (int i = 0; i < 8; ++i) Hb[(size_t)3 * 16 * Cout + i] = acc3[i];
}

// ---------------------------------------------------------------------------
// Kernel 2a: per-channel partial sums over row chunks (coalesced: one thread
// per channel, rows strided across blocks).  Ht:(rows, C)
// ---------------------------------------------------------------------------
__global__ void bn_partial(const float* __restrict__ Ht,
                           float* __restrict__ ps, float* __restrict__ pq,
                           int rows, int C)
{
    const int c = threadIdx.x;                  // blockDim.x == C == 256
    float s = 0.f, q = 0.f;
    for (int r = blockIdx.x; r < rows; r += gridDim.x) {
        const float v = Ht[(size_t)r * C + c];
        s += v; q += v * v;
    }
    ps[(size_t)blockIdx.x * C + c] = s;
    pq[(size_t)blockIdx.x * C + c] = q;
}

// ---------------------------------------------------------------------------
// Kernel 2b: finalize batch stats -> fused affine scale/bias per channel
// ---------------------------------------------------------------------------
__global__ void bn_finalize(const float* __restrict__ ps,
                            const float* __restrict__ pq,
                            const float* __restrict__ gamma,
                            const float* __restrict__ beta,
                            float* __restrict__ scale,
                            float* __restrict__ bias,
                            int G, int rows, int C)
{
    const int c = threadIdx.x;                  // single block, C threads
    float s = 0.f, q = 0.f;
    for (int g = 0; g < G; ++g) {
        s += ps[(size_t)g * C + c];
        q += pq[(size_t)g * C + c];
    }
    const float inv  = 1.0f / (float)rows;
    const float mean = s * inv;
    const float var  = q * inv - mean * mean;
    const float sc   = gamma[c] * rsqrtf(var + BN_EPS);
    scale[c] = sc;
    bias[c]  = beta[c] - mean * sc;
}

// ---------------------------------------------------------------------------
// Kernel 3: three_nn + three_interpolate + lateral BN/ReLU + add.
// One wave per query m; p1[b] staged in LDS (24 KB) via async-to-LDS copies.
// h1t:(B,N,C), h2t:(B,M,C) pre-BN (transposed) -> all gathers coalesced.
// ---------------------------------------------------------------------------
__device__ __forceinline__ void ins3(float d, int i,
                                     float& d0, int& i0,
                                     float& d1, int& i1,
                                     float& d2, int& i2)
{
    if (d < d2) {
        if (d < d1) {
            if (d < d0) { d2 = d1; i2 = i1; d1 = d0; i1 = i0; d0 = d; i0 = i; }
            else        { d2 = d1; i2 = i1; d1 = d;  i1 = i; }
        } else          { d2 = d;  i2 = i; }
    }
}

#define NQ 8   // queries (waves) per block

__global__ void knn_interp(const float* __restrict__ p1,   // (B,N,3)
                           const float* __restrict__ p2,   // (B,M,3)
                           const float* __restrict__ h1t,  // (B,N,C) pre-BN
                           const float* __restrict__ h2t,  // (B,M,C) pre-BN
                           const float* __restrict__ s1, const float* __restrict__ b1,
                           const float* __restrict__ s2, const float* __restrict__ b2,
                           float* __restrict__ y,          // (B,M,C)
                           float* __restrict__ p2o)        // (B,M,3)
{
    __shared__ __align__(16) float sp[NN * 3];   // 24 KB of 320 KB WGP LDS
    const int b = blockIdx.y;
    const float* p1b = p1 + (size_t)b * NN * 3;

#if HAVE_ASYNC_LDS
    {
        // 24 KB = 1536 x 16B chunks, async DMA into LDS (ASYNCcnt path)
        const int chunks = (NN * 3 * 4) / 16;
        gvec_p g = (gvec_p)p1b;                 // global 16B-vector pointer
        lvec_p l = (lvec_p)sp;                  // LDS 16B-vector pointer
        for (int i = threadIdx.x; i < chunks; i += blockDim.x)
            __builtin_amdgcn_global_load_async_to_lds_b128(g + i, l + i, 0, 0);
#if __has_builtin(__builtin_amdgcn_s_wait_asynccnt)
        __builtin_amdgcn_s_wait_asynccnt(0);
#else
        asm volatile("s_wait_asynccnt 0x0" ::: "memory");
#endif
    }
#else
    for (int i = threadIdx.x; i < NN * 3; i += blockDim.x)
        sp[i] = p1b[i];
#endif
    __syncthreads();

    const int wave = threadIdx.x >> 5;
    const int lane = threadIdx.x & 31;
    const int m    = blockIdx.x * NQ + wave;

    const float* q  = p2 + ((size_t)b * MM + m) * 3;
    const float qx = q[0], qy = q[1], qz = q[2];

    float d0 = 3.4e38f, d1 = 3.4e38f, d2 = 3.4e38f;
    int   i0 = 0, i1 = 0, i2 = 0;
    for (int i = lane; i < NN; i += 32) {
        const float dx = sp[i * 3 + 0] - qx;
        const float dy = sp[i * 3 + 1] - qy;
        const float dz = sp[i * 3 + 2] - qz;
        const float d  = dx * dx + dy * dy + dz * dz;
        ins3(d, i, d0, i0, d1, i1, d2, i2);
    }
    // wave32 top-3 merge
    for (int off = 16; off >= 1; off >>= 1) {
        float e0 = __shfl_down(d0, off, 32);
        float e1 = __shfl_down(d1, off, 32);
        float e2 = __shfl_down(d2, off, 32);
        int   j0 = __shfl_down(i0, off, 32);
        int   j1 = __shfl_down(i1, off, 32);
        int   j2 = __shfl_down(i2, off, 32);
        ins3(e0, j0, d0, i0, d1, i1, d2, i2);
        ins3(e1, j1, d0, i0, d1, i1, d2, i2);
        ins3(e2, j2, d0, i0, d1, i1, d2, i2);
    }
    d0 = __shfl(d0, 0, 32); d1 = __shfl(d1, 0, 32); d2 = __shfl(d2, 0, 32);
    i0 = __shfl(i0, 0, 32); i1 = __shfl(i1, 0, 32); i2 = __shfl(i2, 0, 32);

    const float r0 = 1.0f / (d0 + NN_EPS);
    const float r1 = 1.0f / (d1 + NN_EPS);
    const float r2 = 1.0f / (d2 + NN_EPS);
    const float ri = 1.0f / (r0 + r1 + r2);
    const float w0 = r0 * ri, w1 = r1 * ri, w2 = r2 * ri;

    // All rows below are contiguous 256-float vectors -> coalesced per wave
    const float* g0 = h1t + ((size_t)b * NN + i0) * COUT;
    const float* g1 = h1t + ((size_t)b * NN + i1) * COUT;
    const float* g2 = h1t + ((size_t)b * NN + i2) * COUT;
    const float* lt = h2t + ((size_t)b * MM + m) * COUT;
    float*       yb = y   + ((size_t)b * MM + m) * COUT;

#pragma unroll
    for (int o = lane; o < COUT; o += 32) {
        const float sc = s1[o], bi = b1[o];
        const float a0 = fmaxf(0.f, g0[o] * sc + bi);
        const float a1 = fmaxf(0.f, g1[o] * sc + bi);
        const float a2 = fmaxf(0.f, g2[o] * sc + bi);
        const float ll = fmaxf(0.f, lt[o] * s2[o] + b2[o]);
        yb[o] = w0 * a0 + w1 * a1 + w2 * a2 + ll;
    }
    if (lane < 3)
        p2o[((size_t)b * MM + m) * 3 + lane] = q[lane];
}

// ---------------------------------------------------------------------------
// Launch
// ---------------------------------------------------------------------------
extern "C" void kernel_launch(void* const* d_in, const int* in_sizes, int n_in,
                              void* d_out, int out_size, void* d_ws, size_t ws_size,
                              hipStream_t stream)
{
    const float* x1    = (const float*)d_in[0];  // (B,N,Cin)
    const float* p1    = (const float*)d_in[1];  // (B,N,3)
    const float* x2    = (const float*)d_in[2];  // (B,M,Cout)
    const float* p2    = (const float*)d_in[3];  // (B,M,3)
    const float* W_up  = (const float*)d_in[4];  // (Cout,Cin)
    const float* g_up  = (const float*)d_in[5];
    const float* be_up = (const float*)d_in[6];
    const float* W_lat = (const float*)d_in[7];  // (Cout,Cout)
    const float* g_lat = (const float*)d_in[8];
    const float* be_lat= (const float*)d_in[9];

    // workspace layout
    float* h1t = (float*)d_ws;                          // (B,N,COUT)
    float* h2t = h1t + (size_t)BB * NN * COUT;          // (B,M,COUT)
    float* ps  = h2t + (size_t)BB * MM * COUT;          // 256*COUT partial sums
    float* pq  = ps + (size_t)256 * COUT;               // 256*COUT partial sumsq
    float* s1  = pq + (size_t)256 * COUT;
    float* b1  = s1 + COUT;
    float* s2  = b1 + COUT;
    float* b2  = s2 + COUT;

    float* y   = (float*)d_out;                         // (B,M,COUT)
    float* p2o = y + (size_t)BB * MM * COUT;            // (B,M,3)

    // GEMM 1: up_mlp pre-activation.  waves = 2*16*(2048/64) = 1024
    {
        const int waves = BB * (COUT / 16) * (NN / 64);
        gemm_wmma<CIN><<<waves / 4, 128, 0, stream>>>(x1, W_up, h1t, NN, COUT);
    }
    // GEMM 2: lateral_mlp pre-activation. waves = 2*16*(8192/64) = 4096
    {
        const int waves = BB * (COUT / 16) * (MM / 64);
        gemm_wmma<COUT><<<waves / 4, 128, 0, stream>>>(x2, W_lat, h2t, MM, COUT);
    }
    // BN batch statistics -> fused affine (two-stage, deterministic)
    bn_partial<<<256, COUT, 0, stream>>>(h1t, ps, pq, BB * NN, COUT);
    bn_finalize<<<1, COUT, 0, stream>>>(ps, pq, g_up, be_up, s1, b1,
                                        256, BB * NN, COUT);
    bn_partial<<<256, COUT, 0, stream>>>(h2t, ps, pq, BB * MM, COUT);
    bn_finalize<<<1, COUT, 0, stream>>>(ps, pq, g_lat, be_lat, s2, b2,
                                        256, BB * MM, COUT);

    // three_nn + interpolate + lateral + add + p2 passthrough
    dim3 grid(MM / NQ, BB);
    knn_interp<<<grid, NQ * 32, 0, stream>>>(p1, p2, h1t, h2t,
                                             s1, b1, s2, b2, y, p2o);
}